// GATConv_62182536511750
// MI455X (gfx1250) — compile-verified
//
#include <hip/hip_runtime.h>
#include <hip/hip_bf16.h>

#define N_NODES 50000
#define N_EDGES 1600000
#define D_IN    128
#define D_OUT   32
#define ALPHA   0.2f

typedef __attribute__((ext_vector_type(2))) float v2f;
typedef __attribute__((ext_vector_type(8))) float v8f;

// ---------- monotone float <-> ordered-uint bijection for integer min/max atomics ----------
__device__ __forceinline__ unsigned f2ord(float f) {
  unsigned u = __float_as_uint(f);
  return (u & 0x80000000u) ? ~u : (u | 0x80000000u);
}
__device__ __forceinline__ float ord2f(unsigned u) {
  u = (u & 0x80000000u) ? (u & 0x7FFFFFFFu) : ~u;
  return __uint_as_float(u);
}

// ---------- 0: init output / rows_sum / minmax slots ----------
__global__ void gat_init(float* __restrict__ hout, float* __restrict__ rows_sum,
                         unsigned* __restrict__ minmax) {
  int i = blockIdx.x * blockDim.x + threadIdx.x;
  if (i < N_NODES * D_OUT) hout[i] = 0.0f;
  if (i < N_NODES) rows_sum[i] = 0.0f;
  if (i == 0) { minmax[0] = 0xFFFFFFFFu; minmax[1] = 0u; }
}

// ---------- 1: X' = X @ W via V_WMMA_F32_16X16X4_F32 ----------
// One wave -> one 16x32 output tile (two 16x16 accumulators). 50000/16 = 3125 waves.
__global__ void gat_gemm_wmma(const float* __restrict__ X,
                              const float* __restrict__ W,
                              float* __restrict__ Xp) {
  const int wave_in_block = threadIdx.x >> 5;
  const int lane          = threadIdx.x & 31;
  const int wave_id       = blockIdx.x * (blockDim.x >> 5) + wave_in_block;
  const int row_base      = wave_id * 16;
  if (row_base >= N_NODES) return;

  const int m    = lane & 15;   // M for A, N for B/C
  const int kgrp = lane >> 4;   // 0: K=0,1   1: K=2,3  (within each 16x4 step)

  const float* __restrict__ xrow = X + (size_t)(row_base + m) * D_IN;

  v8f acc0 = {};  // N = 0..15
  v8f acc1 = {};  // N = 16..31

  for (int k = 0; k < D_IN; k += 4) {
    const int kk = k + kgrp * 2;
    // A fragment: 16x4 f32, lane (kgrp*16+m) holds K = kgrp*2 + {0,1} of row m
    v2f a;
    a.x = xrow[kk + 0];
    a.y = xrow[kk + 1];
    // B fragments: 4x16 f32, lane (kgrp*16+n) holds K = kgrp*2 + {0,1} of col n
    v2f b0, b1;
    b0.x = W[(size_t)(kk + 0) * D_OUT + m];
    b0.y = W[(size_t)(kk + 1) * D_OUT + m];
    b1.x = W[(size_t)(kk + 0) * D_OUT + 16 + m];
    b1.y = W[(size_t)(kk + 1) * D_OUT + 16 + m];

    acc0 = __builtin_amdgcn_wmma_f32_16x16x4_f32(false, a, false, b0, (short)0, acc0, false, false);
    acc1 = __builtin_amdgcn_wmma_f32_16x16x4_f32(false, a, false, b1, (short)0, acc1, false, false);
  }

  // C/D layout: VGPR i, lanes 0-15 -> row i, lanes 16-31 -> row i+8; col = lane%16
  const int rbase = row_base + kgrp * 8;
  for (int i = 0; i < 8; ++i) {
    const size_t r = (size_t)(rbase + i);
    Xp[r * D_OUT + m]      = acc0[i];
    Xp[r * D_OUT + 16 + m] = acc1[i];
  }
}

// ---------- 2: per-node attention partial scores s0/s1 ----------
__global__ void gat_node_scores(const float* __restrict__ Xp,
                                const float* __restrict__ a0,
                                const float* __restrict__ a1,
                                float* __restrict__ s0,
                                float* __restrict__ s1) {
  int n = blockIdx.x * blockDim.x + threadIdx.x;
  if (n >= N_NODES) return;
  float acc0 = 0.0f, acc1 = 0.0f;
  const float* __restrict__ xr = Xp + (size_t)n * D_OUT;
#pragma unroll
  for (int d = 0; d < D_OUT; ++d) {
    float v = xr[d];
    acc0 += v * a0[d];
    acc1 += v * a1[d];
  }
  s0[n] = acc0;
  s1[n] = acc1;
}

// ---------- 3: edge scores + global min/max (LDS atomics -> 1 global atomic/block) ----------
__global__ void gat_edge_scores(const int* __restrict__ row_idx,
                                const int* __restrict__ col_idx,
                                const float* __restrict__ s0,
                                const float* __restrict__ s1,
                                float* __restrict__ att_raw,
                                unsigned* __restrict__ minmax) {
  __shared__ unsigned bmin, bmax;
  if (threadIdx.x == 0) { bmin = 0xFFFFFFFFu; bmax = 0u; }
  __syncthreads();

  int e = blockIdx.x * blockDim.x + threadIdx.x;
  if (e < N_EDGES) {
    float v = s0[row_idx[e]] + s1[col_idx[e]];
    v = (v > 0.0f) ? v : (ALPHA * v);          // leaky_relu
    att_raw[e] = v;
    unsigned o = f2ord(v);
    atomicMin(&bmin, o);
    atomicMax(&bmax, o);
  }
  __syncthreads();
  if (threadIdx.x == 0) {
    atomicMin(&minmax[0], bmin);
    atomicMax(&minmax[1], bmax);
  }
}

// ---------- 4: normalize+exp, scatter-add h and rows_sum (1 wave per edge, 1 lane per channel) ----------
__global__ void gat_edge_scatter(const int* __restrict__ row_idx,
                                 const int* __restrict__ col_idx,
                                 const float* __restrict__ att_raw,
                                 const float* __restrict__ Xp,
                                 const unsigned* __restrict__ minmax,
                                 float* __restrict__ hout,
                                 float* __restrict__ rows_sum) {
  const int wave = (blockIdx.x * blockDim.x + threadIdx.x) >> 5;
  const int lane = threadIdx.x & 31;
  if (wave >= N_EDGES) return;

  const float vmin = ord2f(minmax[0]);
  const float vmax = ord2f(minmax[1]);
  const float inv  = 1.0f / (vmax - vmin);

  const int r = row_idx[wave];
  const int c = col_idx[wave];
  const float att = __expf((att_raw[wave] - vmin) * inv);

  const float v = att * Xp[(size_t)c * D_OUT + lane];
  atomicAdd(&hout[(size_t)r * D_OUT + lane], v);
  if (lane == 0) atomicAdd(&rows_sum[r], att);
}

// ---------- 5: h /= rows_sum ----------
__global__ void gat_finalize(float* __restrict__ hout,
                             const float* __restrict__ rows_sum) {
  int i = blockIdx.x * blockDim.x + threadIdx.x;
  if (i >= N_NODES * D_OUT) return;
  hout[i] /= rows_sum[i / D_OUT];
}

static inline int cdiv(int a, int b) { return (a + b - 1) / b; }

extern "C" void kernel_launch(void* const* d_in, const int* in_sizes, int n_in,
                              void* d_out, int out_size, void* d_ws, size_t ws_size,
                              hipStream_t stream) {
  (void)in_sizes; (void)n_in; (void)out_size; (void)ws_size;

  const float* X       = (const float*)d_in[0];   // [N_NODES, D_IN]
  const float* W       = (const float*)d_in[1];   // [D_IN, D_OUT]
  const float* a0      = (const float*)d_in[2];   // [1, D_OUT]
  const float* a1      = (const float*)d_in[3];   // [1, D_OUT]
  const int*   row_idx = (const int*)d_in[4];     // [N_EDGES]
  const int*   col_idx = (const int*)d_in[5];     // [N_EDGES]
  float* hout = (float*)d_out;                    // [N_NODES, D_OUT]

  // ---- workspace layout (256B aligned slabs) ----
  char* ws = (char*)d_ws;
  size_t off = 0;
  auto alloc = [&](size_t bytes) { char* p = ws + off; off = (off + bytes + 255) & ~(size_t)255; return p; };
  float*    Xp       = (float*)alloc((size_t)N_NODES * D_OUT * sizeof(float)); // 6.4 MB
  float*    s0       = (float*)alloc((size_t)N_NODES * sizeof(float));
  float*    s1       = (float*)alloc((size_t)N_NODES * sizeof(float));
  float*    att_raw  = (float*)alloc((size_t)N_EDGES * sizeof(float));         // 6.4 MB
  float*    rows_sum = (float*)alloc((size_t)N_NODES * sizeof(float));
  unsigned* minmax   = (unsigned*)alloc(2 * sizeof(unsigned));

  const int TB = 256;

  // 0: init accumulators
  gat_init<<<cdiv(N_NODES * D_OUT, TB), TB, 0, stream>>>(hout, rows_sum, minmax);

  // 1: X' = X @ W (f32 WMMA). 3125 row-tiles of 16; 8 waves per block.
  {
    const int n_waves  = N_NODES / 16;          // 3125
    const int waves_pb = TB / 32;               // 8
    gat_gemm_wmma<<<cdiv(n_waves, waves_pb), TB, 0, stream>>>(X, W, Xp);
  }

  // 2: node scores
  gat_node_scores<<<cdiv(N_NODES, TB), TB, 0, stream>>>(Xp, a0, a1, s0, s1);

  // 3: edge scores + global min/max
  gat_edge_scores<<<cdiv(N_EDGES, TB), TB, 0, stream>>>(row_idx, col_idx, s0, s1, att_raw, minmax);

  // 4: scatter (one wave per edge, one lane per channel)
  {
    const int waves_pb = TB / 32;               // 8 edges per block
    gat_edge_scatter<<<cdiv(N_EDGES, waves_pb), TB, 0, stream>>>(
        row_idx, col_idx, att_raw, Xp, minmax, hout, rows_sum);
  }

  // 5: normalize by per-row sums
  gat_finalize<<<cdiv(N_NODES * D_OUT, TB), TB, 0, stream>>>(hout, rows_sum);
}